// CosineSimilarityLoss_65781719105714
// MI455X (gfx1250) — compile-verified
//
#include <hip/hip_runtime.h>
#include <math.h>

typedef __attribute__((ext_vector_type(2))) float v2f;
typedef __attribute__((ext_vector_type(8))) float v8f;

#define WAVES_PER_BLOCK 4
#define BATCH_PER_WAVE  8                  // 16 rows (2 slots/batch) per wave
#define DIMK            1024
#define CHUNK           32
#define NCHUNK          (DIMK / CHUNK)     // 32
#define LDS_STRIDE      36                 // 32 + 4 pad: 16B-aligned rows, conflict-free columns
#define TILE_FLOATS     (16 * LDS_STRIDE)  // 576 floats per 16x32 tile
#define ROWSTEP_BYTES   (4 * LDS_STRIDE * 4) // 4-row step for the async copies = 576 B

// CDNA5 async LDS copy: each lane copies 16 bytes from the global address in vaddr to the LDS address in vdst, ASYNCcnt-tracked.
#define ASYNC_CP_B128(ldsaddr_u32, gaddr_u64)                                   \
    asm volatile("global_load_async_to_lds_b128 %0, %1, off"                    \
                 :: "v"(ldsaddr_u32), "v"(gaddr_u64) : "memory")

#define WAIT_ASYNC(n)  asm volatile("s_wait_asynccnt " #n ::: "memory")
#define WAIT_DS0()     asm volatile("s_wait_dscnt 0" ::: "memory")

__global__ void zero_out_kernel(float* out) { out[0] = 0.0f; }

__global__ __launch_bounds__(WAVES_PER_BLOCK * 32)
void CosineSimLoss_wmma_kernel(const float* __restrict__ pred,
                               const float* __restrict__ gt,
                               float* __restrict__ out, int B) {
    // Per wave: 4 wave-private tiles: buf0P, buf0G, buf1P, buf1G (double buffered)
    __shared__ float lds[WAVES_PER_BLOCK * 4 * TILE_FLOATS];

    const int lane = threadIdx.x & 31;
    const int wave = threadIdx.x >> 5;

    float* Pl0 = &lds[wave * 4 * TILE_FLOATS];
    float* Gl0 = Pl0 + TILE_FLOATS;
    float* Pl1 = Gl0 + TILE_FLOATS;
    float* Gl1 = Pl1 + TILE_FLOATS;

    // This wave's 16 rows (8 batches x 2 slots) are contiguous in memory.
    const int waveBatch0 = (blockIdx.x * WAVES_PER_BLOCK + wave) * BATCH_PER_WAVE;
    const size_t base = (size_t)waveBatch0 * 2 * DIMK;

    // Async-copy lane mapping: 16 rows x 32 cols, b128 per lane:
    // row = (lane>>3) + 4*i  (i = 0..3), col = (lane&7)*4
    const int rowL = lane >> 3;
    const int colL = (lane & 7) * 4;
    const float* gp = pred + base + (size_t)rowL * DIMK + colL;
    const float* gg = gt   + base + (size_t)rowL * DIMK + colL;

    const unsigned ldsLane = (unsigned)((rowL * LDS_STRIDE + colL) * 4); // byte offset in tile
    const unsigned ldsP0 = (unsigned)(unsigned long long)Pl0 + ldsLane;
    const unsigned ldsG0 = (unsigned)(unsigned long long)Gl0 + ldsLane;
    const unsigned ldsP1 = (unsigned)(unsigned long long)Pl1 + ldsLane;
    const unsigned ldsG1 = (unsigned)(unsigned long long)Gl1 + ldsLane;

    v8f cpg = {};  // P x G^T  -> cross dots
    v8f cpp = {};  // P x P^T  -> diag = |p|^2
    v8f cgg = {};  // G x G^T  -> diag = |g|^2

    // WMMA f32 16x16x4 A-fragment (== B-fragment of the transposed source):
    // lane L holds row (L&15), K = 2*(L>>4) + {0,1}
    const int m = lane & 15;
    const int h = (lane >> 4) * 2;

    // Prologue: fill buffer 0 with chunk 0 (8 async b128 per lane).
    #pragma unroll
    for (int i = 0; i < 4; ++i) {
        ASYNC_CP_B128(ldsP0 + i * ROWSTEP_BYTES,
                      (unsigned long long)(gp + (size_t)i * 4 * DIMK));
        ASYNC_CP_B128(ldsG0 + i * ROWSTEP_BYTES,
                      (unsigned long long)(gg + (size_t)i * 4 * DIMK));
    }
    gp += CHUNK; gg += CHUNK;

    for (int c = 0; c < NCHUNK; ++c) {
        const bool odd = (c & 1);
        const float* Pc = odd ? Pl1 : Pl0;
        const float* Gc = odd ? Gl1 : Gl0;

        if (c + 1 < NCHUNK) {
            // Next chunk goes into the *other* buffer; make sure its previous
            // fragment reads (chunk c-1) have fully drained first.
            WAIT_DS0();
            const unsigned lp = odd ? ldsP0 : ldsP1;
            const unsigned lg = odd ? ldsG0 : ldsG1;
            #pragma unroll
            for (int i = 0; i < 4; ++i) {
                ASYNC_CP_B128(lp + i * ROWSTEP_BYTES,
                              (unsigned long long)(gp + (size_t)i * 4 * DIMK));
                ASYNC_CP_B128(lg + i * ROWSTEP_BYTES,
                              (unsigned long long)(gg + (size_t)i * 4 * DIMK));
            }
            gp += CHUNK; gg += CHUNK;
            // <=8 outstanding: chunk c's 8 copies are done (async loads complete in order);
            // chunk c+1's copies overlap with the WMMA work below.
            WAIT_ASYNC(8);
        } else {
            WAIT_ASYNC(0);
        }

        #pragma unroll
        for (int kk = 0; kk < CHUNK; kk += 4) {
            v2f fp, fg;
            fp.x = Pc[m * LDS_STRIDE + kk + h];
            fp.y = Pc[m * LDS_STRIDE + kk + h + 1];
            fg.x = Gc[m * LDS_STRIDE + kk + h];
            fg.y = Gc[m * LDS_STRIDE + kk + h + 1];
            cpg = __builtin_amdgcn_wmma_f32_16x16x4_f32(false, fp, false, fg,
                                                        (short)0, cpg, false, false);
            cpp = __builtin_amdgcn_wmma_f32_16x16x4_f32(false, fp, false, fp,
                                                        (short)0, cpp, false, false);
            cgg = __builtin_amdgcn_wmma_f32_16x16x4_f32(false, fg, false, fg,
                                                        (short)0, cgg, false, false);
        }
    }

    __syncthreads();

    // Dump accumulators to LDS (wave-private, reuse tiles; 3x256 floats).
    // C layout: VGPR v, lanes 0-15: M=v, N=lane ; lanes 16-31: M=v+8, N=lane-16.
    float* Cpg = Pl0;
    float* Cpp = Gl0;
    float* Cgg = Pl1;
    {
        const int mhalf = (lane >> 4) * 8;
        const int n = lane & 15;
        #pragma unroll
        for (int v = 0; v < 8; ++v) {
            const int M = v + mhalf;
            Cpg[M * 16 + n] = cpg[v];
            Cpp[M * 16 + n] = cpp[v];
            Cgg[M * 16 + n] = cgg[v];
        }
    }
    __syncthreads();

    float loss = 0.0f;
    if (lane < BATCH_PER_WAVE) {
        const int r0 = 2 * lane, r1 = 2 * lane + 1;
        const float pp0 = Cpp[r0 * 16 + r0], pp1 = Cpp[r1 * 16 + r1];
        const float gg0 = Cgg[r0 * 16 + r0], gg1 = Cgg[r1 * 16 + r1];
        const float pg00 = Cpg[r0 * 16 + r0], pg01 = Cpg[r0 * 16 + r1];
        const float pg10 = Cpg[r1 * 16 + r0], pg11 = Cpg[r1 * 16 + r1];
        const float eps = 1e-12f;
        const float ip0 = 1.0f / fmaxf(sqrtf(pp0), eps);
        const float ip1 = 1.0f / fmaxf(sqrtf(pp1), eps);
        const float ig0 = 1.0f / fmaxf(sqrtf(gg0), eps);
        const float ig1 = 1.0f / fmaxf(sqrtf(gg1), eps);
        const float c00 = pg00 * ip0 * ig0;
        const float c01 = pg01 * ip0 * ig1;
        const float c10 = pg10 * ip1 * ig0;
        const float c11 = pg11 * ip1 * ig1;
        const float best = fmaxf(c00 + c11, c01 + c10);
        loss = 1.0f - 0.5f * best;
    }
    loss += __shfl_xor(loss, 1, 32);
    loss += __shfl_xor(loss, 2, 32);
    loss += __shfl_xor(loss, 4, 32);
    if (lane == 0) atomicAdd(out, loss / (float)B);
}

extern "C" void kernel_launch(void* const* d_in, const int* in_sizes, int n_in,
                              void* d_out, int out_size, void* d_ws, size_t ws_size,
                              hipStream_t stream) {
    const float* pred = (const float*)d_in[0];
    const float* gt   = (const float*)d_in[1];
    float* out        = (float*)d_out;

    const int B = in_sizes[0] / (2 * DIMK);   // 32768

    zero_out_kernel<<<1, 1, 0, stream>>>(out);

    const int batchesPerBlock = WAVES_PER_BLOCK * BATCH_PER_WAVE;  // 32
    const int grid = (B + batchesPerBlock - 1) / batchesPerBlock;  // 1024
    CosineSimLoss_wmma_kernel<<<grid, WAVES_PER_BLOCK * 32, 0, stream>>>(pred, gt, out, B);
}